// RoIAlign_3470333575868
// MI455X (gfx1250) — compile-verified
//
#include <hip/hip_runtime.h>

// TF crop_and_resize (bilinear, corner-aligned, extrapolation_value = 1.0)
// images: [B,256,256,256] f32, boxes: [B,100,4] f32 -> out: [B,7,7,100,256] f32
//
// Memory-bound gather kernel (arith intensity ~0.45 flop/byte): no matrix
// structure (corner data is position-dependent, so WMMA's shared-B form does
// not apply). Optimized for MI455X memory system: coalesced B128 gathers along
// the contiguous channel axis, SALU-scalarized per-position math (3D grid to
// minimize the scalar index-decode chain before loads can issue), and
// non-temporal (TH=NT) output stores to preserve the 192MB L2 for the
// 4x-overlapping image corner reads.

typedef float v4f __attribute__((ext_vector_type(4)));

#define HH   256
#define WW   256
#define CC   256
#define NBOX 100
#define CROP 7

__global__ __launch_bounds__(64) void crop_resize_k(
    const float* __restrict__ img,
    const float* __restrict__ boxes,
    float* __restrict__ out)
{
    // grid: (x = n, y = i*7+j, z = b) -> output layout [b,i,j,n,c]
    int n  = blockIdx.x;
    int ij = blockIdx.y;
    int b  = blockIdx.z;
    int i  = ij / CROP;
    int j  = ij % CROP;

    // Uniform (blockIdx-only) -> compiler scalarizes to s_load + SALU.
    const float* bp = boxes + (((size_t)b * NBOX + n) << 2);
    float y1 = bp[0], x1 = bp[1], y2 = bp[2], x2 = bp[3];

    // corner-aligned sample grid (crop_size > 1 branch of the TF kernel)
    float in_y = y1 * (float)(HH - 1)
               + (float)i * ((y2 - y1) * (float)(HH - 1) / (float)(CROP - 1));
    float in_x = x1 * (float)(WW - 1)
               + (float)j * ((x2 - x1) * (float)(WW - 1) / (float)(CROP - 1));

    size_t pos = ((size_t)b * (CROP * CROP) + ij) * NBOX + n;
    v4f* op = reinterpret_cast<v4f*>(out + pos * CC) + threadIdx.x;

    bool valid = (in_y >= 0.0f) && (in_y <= (float)(HH - 1)) &&
                 (in_x >= 0.0f) && (in_x <= (float)(WW - 1));
    if (!valid) {                       // uniform across the block
        v4f one = {1.0f, 1.0f, 1.0f, 1.0f};
        __builtin_nontemporal_store(one, op);
        return;
    }

    float y0f = floorf(in_y);
    float x0f = floorf(in_x);
    float ly  = in_y - y0f;
    float lx  = in_x - x0f;
    int yt = (int)fminf(fmaxf(y0f,         0.0f), (float)(HH - 1));
    int yb = (int)fminf(fmaxf(ceilf(in_y), 0.0f), (float)(HH - 1));
    int xl = (int)fminf(fmaxf(x0f,         0.0f), (float)(WW - 1));
    int xr = (int)fminf(fmaxf(ceilf(in_x), 0.0f), (float)(WW - 1));

    // Corner row segments: 1KB contiguous per corner, coalesced across 64 lanes.
    size_t base = (size_t)b * ((size_t)HH * WW * CC);
    const v4f* r00 = reinterpret_cast<const v4f*>(img + base + ((size_t)yt * WW + xl) * CC) + threadIdx.x;
    const v4f* r01 = reinterpret_cast<const v4f*>(img + base + ((size_t)yt * WW + xr) * CC) + threadIdx.x;
    const v4f* r10 = reinterpret_cast<const v4f*>(img + base + ((size_t)yb * WW + xl) * CC) + threadIdx.x;
    const v4f* r11 = reinterpret_cast<const v4f*>(img + base + ((size_t)yb * WW + xr) * CC) + threadIdx.x;

    v4f p00 = *r00;
    v4f p01 = *r01;
    v4f p10 = *r10;
    v4f p11 = *r11;

    float omlx = 1.0f - lx;
    float omly = 1.0f - ly;
    v4f top = p00 * omlx + p01 * lx;
    v4f bot = p10 * omlx + p11 * lx;
    v4f val = top * omly + bot * ly;

    // Output is write-once (40MB): stream it past L2 (TH=NT) so gathered image
    // lines stay resident.
    __builtin_nontemporal_store(val, op);
}

extern "C" void kernel_launch(void* const* d_in, const int* in_sizes, int n_in,
                              void* d_out, int out_size, void* d_ws, size_t ws_size,
                              hipStream_t stream) {
    const float* img   = (const float*)d_in[0];
    const float* boxes = (const float*)d_in[1];
    float* out = (float*)d_out;

    int B = in_sizes[1] / (NBOX * 4);          // boxes: [B,100,4]
    dim3 grid(NBOX, CROP * CROP, B);           // one 64-thread block per sample position
    crop_resize_k<<<grid, dim3(64), 0, stream>>>(img, boxes, out);
}